// Model_40364102648388
// MI455X (gfx1250) — compile-verified
//
#include <hip/hip_runtime.h>
#include <hip/hip_bf16.h>
#include <math.h>

typedef __attribute__((ext_vector_type(2))) float v2f;
typedef __attribute__((ext_vector_type(8))) float v8f;

__device__ __forceinline__ float sigmoidf_(float x){ return 1.f/(1.f+__expf(-x)); }
__device__ __forceinline__ float siluf_(float x){ return x*sigmoidf_(x); }
__device__ __forceinline__ float geluf_(float x){ return 0.5f*x*(1.f+erff(x*0.7071067811865475f)); }

// ---------------------------------------------------------------------------
// WMMA fp32 GEMM: C[M,N] = act(A[M,K] @ W[N,K]^T + bias[N])
// 256 threads = 8 waves arranged 2(M) x 4(N); each wave computes a 32x32 tile
// with 2x2 v8f accumulators (fragment reuse: 4 WMMA per 4 LDS pair-loads).
// Block tile 64(M) x 128(N); K staged through LDS in chunks of 32.
// Fragment layouts per CDNA5 ISA 7.12.2 (32-bit):
//   A 16x4 : lane l -> M = l&15 ; vgpr j, half h=l>>4 -> K = 2h+j
//   B 4x16 : lane l -> N = l&15 ; vgpr j, half h=l>>4 -> K = 2h+j
//   C 16x16: vgpr v, lane l -> M = v + 8*(l>>4), N = l&15
// ---------------------------------------------------------------------------
#define TM 64
#define TN 128
#define TKX 32

__global__ __launch_bounds__(256) void gemm_wmma_f32(
    const float* __restrict__ A, const float* __restrict__ W,
    const float* __restrict__ bias, float* __restrict__ C,
    int M, int N, int K, int act)
{
  __shared__ float As[TM][TKX + 1];
  __shared__ float Ws[TN][TKX + 1];
  const int tid  = threadIdx.x;
  const int lane = tid & 31;
  const int wav  = tid >> 5;
  const int bm = blockIdx.x * TM;
  const int bn = blockIdx.y * TN;
  const int wm = (wav >> 2) * 32;   // 0,32
  const int wn = (wav & 3) * 32;    // 0,32,64,96

  v8f acc00 = {}, acc01 = {}, acc10 = {}, acc11 = {};
  const int mrow = wm + (lane & 15);
  const int nrow = wn + (lane & 15);
  const int kh   = (lane >> 4) << 1;

  for (int k0 = 0; k0 < K; k0 += TKX) {
    const int kn = k0 + TKX;          // next chunk (prefetch target)
    for (int i = tid; i < TM * TKX; i += 256) {
      int r = i >> 5, c = i & 31;
      int gm = bm + r, gk = k0 + c;
      As[r][c] = (gm < M && gk < K) ? A[(size_t)gm * K + gk] : 0.f;
      if (c == 0 && gm < M && kn < K)
        __builtin_prefetch(&A[(size_t)gm * K + kn], 0, 1);   // global_prefetch_b8
    }
    for (int i = tid; i < TN * TKX; i += 256) {
      int r = i >> 5, c = i & 31;
      int gn = bn + r, gk = k0 + c;
      Ws[r][c] = (gn < N && gk < K) ? W[(size_t)gn * K + gk] : 0.f;
      if (c == 0 && gn < N && kn < K)
        __builtin_prefetch(&W[(size_t)gn * K + kn], 0, 1);
    }
    __syncthreads();
#pragma unroll
    for (int kk = 0; kk < TKX; kk += 4) {
      v2f a0, a1, b0, b1;
      a0[0] = As[mrow][kk + kh];        a0[1] = As[mrow][kk + kh + 1];
      a1[0] = As[mrow + 16][kk + kh];   a1[1] = As[mrow + 16][kk + kh + 1];
      b0[0] = Ws[nrow][kk + kh];        b0[1] = Ws[nrow][kk + kh + 1];
      b1[0] = Ws[nrow + 16][kk + kh];   b1[1] = Ws[nrow + 16][kk + kh + 1];
      acc00 = __builtin_amdgcn_wmma_f32_16x16x4_f32(false, a0, false, b0, (short)0, acc00, false, false);
      acc01 = __builtin_amdgcn_wmma_f32_16x16x4_f32(false, a0, false, b1, (short)0, acc01, false, false);
      acc10 = __builtin_amdgcn_wmma_f32_16x16x4_f32(false, a1, false, b0, (short)0, acc10, false, false);
      acc11 = __builtin_amdgcn_wmma_f32_16x16x4_f32(false, a1, false, b1, (short)0, acc11, false, false);
    }
    __syncthreads();
  }

  const int mh = (lane >> 4) << 3;
  const int nc = lane & 15;
#pragma unroll
  for (int half = 0; half < 4; ++half) {
    const v8f* accp = (half == 0) ? &acc00 : (half == 1) ? &acc01 : (half == 2) ? &acc10 : &acc11;
    int moff = (half >> 1) * 16;      // acc1x -> M+16
    int noff = (half & 1) * 16;       // accx1 -> N+16
    int gn = bn + wn + noff + nc;
    float bv = 0.f;
    if (bias && gn < N) bv = bias[gn];
#pragma unroll
    for (int v = 0; v < 8; ++v) {
      int gm = bm + wm + moff + mh + v;
      if (gm < M && gn < N) {
        float x = (*accp)[v] + bv;
        if (act == 1)      x = geluf_(x);
        else if (act == 2) x = siluf_(x);
        C[(size_t)gm * N + gn] = x;
      }
    }
  }
}

// ---------------------------------------------------------------------------
// Instance norm over L per (b,n); also stores mean/stdev for final de-norm.
// grid = B, block = NV (128)
// ---------------------------------------------------------------------------
__global__ void instnorm_kernel(const float* __restrict__ x, float* __restrict__ xn,
                                float* __restrict__ mean, float* __restrict__ stdev,
                                int Bb, int Lseq, int Nv)
{
  int b = blockIdx.x, n = threadIdx.x;
  float s = 0.f, s2 = 0.f;
  for (int t = 0; t < Lseq; ++t) {
    float v = x[((size_t)b * Lseq + t) * Nv + n];
    s += v; s2 += v * v;
  }
  float mu = s / Lseq;
  float var = s2 / Lseq - mu * mu;
  float sd = sqrtf(var + 1e-5f);
  mean[b * Nv + n] = mu;
  stdev[b * Nv + n] = sd;
  float inv = 1.f / sd;
  for (int t = 0; t < Lseq; ++t) {
    size_t i = ((size_t)b * Lseq + t) * Nv + n;
    xn[i] = (x[i] - mu) * inv;
  }
}

// depthwise causal conv (4 taps) + SiLU. Input = first Di cols of xz [B,L,2Di].
// rev: emulate conv over the time-reversed sequence; output in reversed layout.
__global__ void conv_silu_kernel(const float* __restrict__ xz,
                                 const float* __restrict__ cw, const float* __restrict__ cb,
                                 float* __restrict__ xi,
                                 int Bb, int Lseq, int Di, int twoDi, int rev)
{
  int idx = blockIdx.x * blockDim.x + threadIdx.x;
  int tot = Bb * Lseq * Di;
  if (idx >= tot) return;
  int c = idx % Di;
  int t = (idx / Di) % Lseq;
  int b = idx / (Di * Lseq);
  float acc = cb[c];
#pragma unroll
  for (int j = 0; j < 4; ++j) {
    int s = t - 3 + j;                 // logical (possibly reversed) time index
    if (s >= 0) {
      int tt = rev ? (Lseq - 1 - s) : s;
      acc += xz[((size_t)b * Lseq + tt) * twoDi + c] * cw[j * Di + c];
    }
  }
  xi[idx] = siluf_(acc);
}

// dt = softplus(dbl[:, :R] @ W_dt^T + b_dt)   (K = 8 or 32 -> plain VALU)
__global__ void dt_softplus_kernel(const float* __restrict__ dbl,
                                   const float* __restrict__ Wdt, const float* __restrict__ bdt,
                                   float* __restrict__ dt,
                                   int Mrows, int Di, int R, int dblStride)
{
  int idx = blockIdx.x * blockDim.x + threadIdx.x;
  if (idx >= Mrows * Di) return;
  int d = idx % Di, m = idx / Di;
  const float* row = dbl + (size_t)m * dblStride;
  float acc = bdt[d];
  for (int r = 0; r < R; ++r) acc += row[r] * Wdt[d * R + r];
  dt[idx] = (acc > 20.f) ? acc : log1pf(__expf(acc));
}

// selective scan: one thread per (b, d), 32 states in registers, sequential in t.
__global__ __launch_bounds__(256) void scan_kernel(
    const float* __restrict__ xi, const float* __restrict__ dt,
    const float* __restrict__ dbl, const float* __restrict__ A_log,
    const float* __restrict__ Dp, float* __restrict__ y,
    int Bb, int Lseq, int Di, int R, int dblStride)
{
  int idx = blockIdx.x * blockDim.x + threadIdx.x;
  if (idx >= Bb * Di) return;
  int d = idx % Di, b = idx / Di;
  float Aa[32], h[32];
#pragma unroll
  for (int s = 0; s < 32; ++s) { Aa[s] = -__expf(A_log[d * 32 + s]); h[s] = 0.f; }
  float Dv = Dp[d];
  for (int t = 0; t < Lseq; ++t) {
    size_t row = (size_t)b * Lseq + t;
    float dtv = dt[row * Di + d];
    float u   = xi[row * Di + d];
    float du  = dtv * u;
    const float* bm = dbl + row * dblStride + R;
    const float* cm = bm + 32;
    float acc = 0.f;
#pragma unroll
    for (int s = 0; s < 32; ++s) {
      float dA = __expf(dtv * Aa[s]);
      h[s] = dA * h[s] + du * bm[s];
      acc += h[s] * cm[s];
    }
    y[row * Di + d] = acc + u * Dv;
  }
}

// y *= silu(z), z = xz[:, Di:2Di] read at (optionally reversed) time.
__global__ void gate_kernel(float* __restrict__ y, const float* __restrict__ xz,
                            int Bb, int Lseq, int Di, int twoDi, int rev)
{
  int idx = blockIdx.x * blockDim.x + threadIdx.x;
  int tot = Bb * Lseq * Di;
  if (idx >= tot) return;
  int d = idx % Di;
  int t = (idx / Di) % Lseq;
  int b = idx / (Di * Lseq);
  int tt = rev ? (Lseq - 1 - t) : t;
  float z = xz[((size_t)b * Lseq + tt) * twoDi + Di + d];
  y[idx] *= siluf_(z);
}

// out = LN(x [+ f] [+ r(reversed t)]) ; one block per row, width <= 512.
__global__ __launch_bounds__(256) void add_ln_kernel(
    const float* __restrict__ x, const float* __restrict__ f, const float* __restrict__ r,
    const float* __restrict__ g, const float* __restrict__ bet, float* __restrict__ out,
    int rows, int width, int Lseq)
{
  __shared__ float vals[512];
  __shared__ float ssum[256];
  __shared__ float ssq[256];
  int row = blockIdx.x, tid = threadIdx.x;
  int bidx = row / Lseq, t = row % Lseq;
  size_t o   = (size_t)row * width;
  size_t orv = ((size_t)bidx * Lseq + (Lseq - 1 - t)) * width;
  float s = 0.f, s2 = 0.f;
  for (int c = tid; c < width; c += 256) {
    float v = x[o + c];
    if (f) v += f[o + c];
    if (r) v += r[orv + c];
    vals[c] = v; s += v; s2 += v * v;
  }
  ssum[tid] = s; ssq[tid] = s2;
  __syncthreads();
  for (int k = 128; k > 0; k >>= 1) {
    if (tid < k) { ssum[tid] += ssum[tid + k]; ssq[tid] += ssq[tid + k]; }
    __syncthreads();
  }
  float mean = ssum[0] / width;
  float var  = ssq[0] / width - mean * mean;
  float rs   = rsqrtf(var + 1e-5f);
  for (int c = tid; c < width; c += 256)
    out[o + c] = (vals[c] - mean) * rs * g[c] + bet[c];
}

// [B,L,N] -> [B,N,L]
__global__ void transpose_kernel(const float* __restrict__ in, float* __restrict__ out,
                                 int Bb, int Lseq, int Nv)
{
  int idx = blockIdx.x * blockDim.x + threadIdx.x;
  int tot = Bb * Lseq * Nv;
  if (idx >= tot) return;
  int l = idx % Lseq;
  int n = (idx / Lseq) % Nv;
  int b = idx / (Lseq * Nv);
  out[idx] = in[((size_t)b * Lseq + l) * Nv + n];
}

// dec[b,p,n] = proj[b*N+n][p] * stdev[b,n] + mean[b,n]
__global__ void final_kernel(const float* __restrict__ po, const float* __restrict__ mean,
                             const float* __restrict__ stdev, float* __restrict__ dec,
                             int Bb, int Nv, int P)
{
  int idx = blockIdx.x * blockDim.x + threadIdx.x;
  int tot = Bb * P * Nv;
  if (idx >= tot) return;
  int n = idx % Nv;
  int p = (idx / Nv) % P;
  int b = idx / (Nv * P);
  dec[idx] = po[((size_t)b * Nv + n) * P + p] * stdev[b * Nv + n] + mean[b * Nv + n];
}

// ---------------------------------------------------------------------------
extern "C" void kernel_launch(void* const* d_in, const int* in_sizes, int n_in,
                              void* d_out, int out_size, void* d_ws, size_t ws_size,
                              hipStream_t stream)
{
  (void)in_sizes; (void)n_in; (void)out_size; (void)ws_size;
  const int Bb = 8, L = 512, NV = 128, DM = 512, P = 96, DFF = 2048;

  const float* x_enc = (const float*)d_in[0];
  int pi = 4;  // skip x_enc, x_mark_enc, x_dec, x_mark_dec
  auto F = [&](int i) -> const float* { return (const float*)d_in[i]; };

  struct MambaP { const float *W_in,*conv_w,*conv_b,*W_x,*W_dt,*b_dt,*A_log,*D,*W_out; };
  auto readM = [&]() -> MambaP {
    MambaP m;
    m.W_in = F(pi++);  m.conv_w = F(pi++); m.conv_b = F(pi++);
    m.W_x  = F(pi++);  m.W_dt   = F(pi++); m.b_dt   = F(pi++);
    m.A_log = F(pi++); m.D      = F(pi++); m.W_out  = F(pi++);
    return m;
  };

  MambaP tf[2], tb[2]; const float *tlng[2], *tlnb[2];
  for (int l = 0; l < 2; ++l) { tf[l] = readM(); tb[l] = readM(); tlng[l] = F(pi++); tlnb[l] = F(pi++); }
  const float *eW1 = F(pi++), *eb1 = F(pi++), *eW2 = F(pi++), *eb2 = F(pi++);
  MambaP att[2], attr[2];
  const float *n1g[2], *n1b[2], *c1w[2], *c1b[2], *c2w[2], *c2b[2], *n2g[2], *n2b[2];
  for (int l = 0; l < 2; ++l) {
    att[l] = readM(); attr[l] = readM();
    n1g[l] = F(pi++); n1b[l] = F(pi++);
    c1w[l] = F(pi++); c1b[l] = F(pi++);
    c2w[l] = F(pi++); c2b[l] = F(pi++);
    n2g[l] = F(pi++); n2b[l] = F(pi++);
  }
  const float *encg = F(pi++), *encb = F(pi++);
  const float *pW = F(pi++), *pb = F(pi++);

  // ---- workspace carve-up (~44 MB) ----
  char* wsb = (char*)d_ws; size_t woff = 0;
  auto alloc = [&](size_t n) -> float* {
    float* p = (float*)(wsb + woff);
    woff += ((n * sizeof(float) + 255) & ~(size_t)255);
    return p;
  };
  float* meanb = alloc(Bb * NV);
  float* stdb  = alloc(Bb * NV);
  float* xa  = alloc((size_t)Bb * L * NV);
  float* xb  = alloc((size_t)Bb * L * NV);
  float* fo  = alloc((size_t)Bb * NV * DM);     // 524288: fits B*L*NV too
  float* ro  = alloc((size_t)Bb * NV * DM);
  float* xz  = alloc((size_t)4096 * 512);
  float* xi  = alloc((size_t)4096 * 256);
  float* dbl = alloc((size_t)4096 * 72);
  float* dtb = alloc((size_t)4096 * 256);
  float* ys  = alloc((size_t)4096 * 256);
  float* et  = alloc((size_t)Bb * NV * L);
  float* ea  = alloc((size_t)Bb * NV * DM);
  float* ebf = alloc((size_t)Bb * NV * DM);
  float* hb  = alloc((size_t)Bb * NV * DFF);
  float* po  = alloc((size_t)Bb * NV * P);

  auto gemm = [&](const float* A_, const float* W_, const float* b_, float* C_,
                  int M, int N, int K, int act) {
    dim3 g((M + TM - 1) / TM, (N + TN - 1) / TN);
    gemm_wmma_f32<<<g, dim3(256), 0, stream>>>(A_, W_, b_, C_, M, N, K, act);
  };

  auto run_mamba = [&](const MambaP& p, const float* in, float* out,
                       int Lseq, int dm, int Di, int R, int rev) {
    int Mr = Bb * Lseq, twoDi = 2 * Di, dblN = R + 64;
    int tot = Mr * Di;
    gemm(in, p.W_in, nullptr, xz, Mr, twoDi, dm, 0);
    conv_silu_kernel<<<(tot + 255) / 256, 256, 0, stream>>>(
        xz, p.conv_w, p.conv_b, xi, Bb, Lseq, Di, twoDi, rev);
    gemm(xi, p.W_x, nullptr, dbl, Mr, dblN, Di, 0);
    dt_softplus_kernel<<<(tot + 255) / 256, 256, 0, stream>>>(
        dbl, p.W_dt, p.b_dt, dtb, Mr, Di, R, dblN);
    int nth = Bb * Di;
    scan_kernel<<<(nth + 255) / 256, 256, 0, stream>>>(
        xi, dtb, dbl, p.A_log, p.D, ys, Bb, Lseq, Di, R, dblN);
    gate_kernel<<<(tot + 255) / 256, 256, 0, stream>>>(
        ys, xz, Bb, Lseq, Di, twoDi, rev);
    gemm(ys, p.W_out, nullptr, out, Mr, dm, Di, 0);
  };

  // ---- 1) instance norm ----
  instnorm_kernel<<<Bb, NV, 0, stream>>>(x_enc, xa, meanb, stdb, Bb, L, NV);

  // ---- 2) temporal bidirectional Mamba layers (d_model=128, Di=256, R=8) ----
  float* xc = xa; float* xn = xb;
  for (int l = 0; l < 2; ++l) {
    run_mamba(tf[l], xc, fo, L, NV, 256, 8, 0);
    run_mamba(tb[l], xc, ro, L, NV, 256, 8, 1);
    add_ln_kernel<<<Bb * L, 256, 0, stream>>>(xc, fo, ro, tlng[l], tlnb[l], xn, Bb * L, NV, L);
    float* t = xc; xc = xn; xn = t;
  }

  // ---- 3) embedding MLP over [B,N,L] ----
  {
    int tot = Bb * NV * L;
    transpose_kernel<<<(tot + 255) / 256, 256, 0, stream>>>(xc, et, Bb, L, NV);
  }
  gemm(et, eW1, eb1, hb, Bb * NV, DM * 2, L, 1);        // GELU
  gemm(hb, eW2, eb2, ea, Bb * NV, DM, DM * 2, 0);

  // ---- 4) encoder layers (seq=N=128, d_model=512, Di=512, R=32) ----
  float* ec = ea; float* en = ebf;
  for (int l = 0; l < 2; ++l) {
    run_mamba(att[l],  ec, fo, NV, DM, DM, 32, 0);
    run_mamba(attr[l], ec, ro, NV, DM, DM, 32, 1);
    add_ln_kernel<<<Bb * NV, 256, 0, stream>>>(ec, fo, ro, n1g[l], n1b[l], en, Bb * NV, DM, NV);
    gemm(en, c1w[l], c1b[l], hb, Bb * NV, DFF, DM, 1);  // GELU
    gemm(hb, c2w[l], c2b[l], fo, Bb * NV, DM, DFF, 0);
    add_ln_kernel<<<Bb * NV, 256, 0, stream>>>(en, fo, nullptr, n2g[l], n2b[l], ec, Bb * NV, DM, NV);
  }

  // ---- 5) final norm, projection, de-norm ----
  add_ln_kernel<<<Bb * NV, 256, 0, stream>>>(ec, nullptr, nullptr, encg, encb, en, Bb * NV, DM, NV);
  gemm(en, pW, pb, po, Bb * NV, P, DM, 0);
  {
    int tot = Bb * P * NV;
    final_kernel<<<(tot + 255) / 256, 256, 0, stream>>>(po, meanb, stdb, (float*)d_out, Bb, NV, P);
  }
}